// TrajsFeatures_39152921870522
// MI455X (gfx1250) — compile-verified
//
#include <hip/hip_runtime.h>
#include <math.h>

typedef float v2f __attribute__((ext_vector_type(2)));
typedef float v8f __attribute__((ext_vector_type(8)));

#define TPTS 256   // points per graph (fixed by the reference)

// ---------------------------------------------------------------------------
// Kernel 1: one 256-thread block per graph.
//  - loads P tile to LDS (coalesced)
//  - block reductions -> pos_std + step stats (6 per-graph scale outputs)
//  - WMMA-based inclusive scan (v_wmma_f32_16x16x4_f32, full fp32) of
//    dr_norm and dr_norm^2 over the 256 points
//  - writes X[n,3] = (time_norm, cum_dist, cum_sd)
// ---------------------------------------------------------------------------
__global__ __launch_bounds__(256) void traj_node_kernel(
    const float* __restrict__ P,        // [N,3]
    float* __restrict__ X,              // [N,3]
    float* __restrict__ scales,         // 6 arrays of [G] (sorted-key order)
    float* __restrict__ posstd_ws,      // [G] scratch for edge kernel
    int Gn)
{
    __shared__ float Pl[TPTS * 3];
    __shared__ float xA[TPTS];          // dr_norm
    __shared__ float xB[TPTS];          // dr_norm^2
    __shared__ float red[8 * 8];        // per-wave partial sums
    __shared__ float tl[2][16];         // row totals for the two scans
    __shared__ float bcast[1];          // pos_std broadcast

    const int g  = blockIdx.x;
    const int t  = threadIdx.x;
    const int n0 = g * TPTS;

    // ---- coalesced stage of this graph's positions into LDS -------------
    for (int i = t; i < TPTS * 3; i += 256)
        Pl[i] = P[(size_t)n0 * 3 + i];
    __syncthreads();

    const float px = Pl[3 * t + 0];
    const float py = Pl[3 * t + 1];
    const float pz = Pl[3 * t + 2];

    float drx = 0.f, dry = 0.f, drz = 0.f;
    if (t < TPTS - 1) {
        drx = Pl[3 * (t + 1) + 0] - px;
        dry = Pl[3 * (t + 1) + 1] - py;
        drz = Pl[3 * (t + 1) + 2] - pz;
    }
    const float dr2 = drx * drx + dry * dry + drz * drz;
    const float dn0 = sqrtf(1e-5f + dr2);          // unscaled step length
    const float inm = (t < TPTS - 1) ? 1.f : 0.f;  // in-points mask

    // ---- block reduction of 8 statistics --------------------------------
    float v[8] = { px, py, pz, px * px, py * py, pz * pz,
                   dn0 * inm, dn0 * dn0 * inm };
    #pragma unroll
    for (int off = 16; off >= 1; off >>= 1)
        #pragma unroll
        for (int k = 0; k < 8; ++k)
            v[k] += __shfl_xor(v[k], off, 32);

    const int lane = t & 31;
    const int wid  = t >> 5;
    if (lane == 0)
        #pragma unroll
        for (int k = 0; k < 8; ++k) red[wid * 8 + k] = v[k];
    __syncthreads();

    if (t == 0) {
        float s[8];
        #pragma unroll
        for (int k = 0; k < 8; ++k) s[k] = 0.f;
        for (int w = 0; w < 8; ++w)
            #pragma unroll
            for (int k = 0; k < 8; ++k) s[k] += red[w * 8 + k];

        const float invL = 1.f / (float)TPTS;
        const float mx = s[0] * invL, my = s[1] * invL, mz = s[2] * invL;
        float var = (s[3] * invL - mx * mx) + (s[4] * invL - my * my)
                  + (s[5] * invL - mz * mz);
        const float pos_std   = sqrtf(var);
        const float cntInv    = 1.f / (float)(TPTS - 1);
        const float step_sum  = s[6];
        const float step_mean = step_sum * cntInv;
        const float step_var  = s[7] * cntInv;
        const float step_std  = sqrtf(step_var - step_mean * step_mean);

        // JAX pytree dict flatten = sorted keys:
        // log_L, pos_std, step_mean, step_std, step_sum, step_var
        scales[0 * (size_t)Gn + g] = logf((float)TPTS);
        scales[1 * (size_t)Gn + g] = pos_std;
        scales[2 * (size_t)Gn + g] = step_mean;
        scales[3 * (size_t)Gn + g] = step_std;
        scales[4 * (size_t)Gn + g] = step_sum;
        scales[5 * (size_t)Gn + g] = step_var;
        posstd_ws[g] = pos_std;
        bcast[0] = pos_std;
    }
    __syncthreads();

    const float sf   = bcast[0];
    const float inv2 = 1.f / (sf * sf);
    const float dn   = sqrtf(1e-5f + dr2 * inv2); // scaled step length
    xA[t] = dn;
    xB[t] = dn * dn;
    X[(size_t)(n0 + t) * 3 + 0] = (float)t * (1.f / 255.f); // time_norm
    __syncthreads();

    // ---- WMMA inclusive scan over 256 elements ---------------------------
    // View x as 16x16 row-major Xm. Y = Xm*U (within-row scans), row totals
    // tr = Y[:,15], R = Y + Lstrict*T (T broadcasts tr). K=4 fp32 WMMAs.
    const int  hh = lane >> 4;    // lane half: K offset +2 for A/B regs
    const int  nn = lane & 15;    // N column for B/C/D; M row for A
    const bool act = (wid < 2);
    const float* xl = (wid == 0) ? xA : xB;

    v8f R = {};
    if (act) {
        v8f Y = {};
        #pragma unroll
        for (int c = 0; c < 4; ++c) {
            const int k0 = 4 * c + 2 * hh;
            v2f A, Bv;
            A.x  = xl[16 * nn + k0];                // Xm[nn][k0]
            A.y  = xl[16 * nn + k0 + 1];
            Bv.x = (k0     <= nn) ? 1.f : 0.f;      // U[k0][nn]
            Bv.y = (k0 + 1 <= nn) ? 1.f : 0.f;
            Y = __builtin_amdgcn_wmma_f32_16x16x4_f32(
                    false, A, false, Bv, (short)0, Y, false, false);
        }
        if (nn == 15)                                // column 15 = row totals
            #pragma unroll
            for (int r = 0; r < 8; ++r)
                tl[wid][r + 8 * hh] = Y[r];
        R = Y;
    }
    __syncthreads();

    if (act) {
        #pragma unroll
        for (int c = 0; c < 4; ++c) {
            const int k0 = 4 * c + 2 * hh;
            v2f A, Bv;
            A.x  = (k0     < nn) ? 1.f : 0.f;       // Lstrict[nn][k0]
            A.y  = (k0 + 1 < nn) ? 1.f : 0.f;
            Bv.x = tl[wid][k0];                     // T[k0][*] = tr[k0]
            Bv.y = tl[wid][k0 + 1];
            R = __builtin_amdgcn_wmma_f32_16x16x4_f32(
                    false, A, false, Bv, (short)0, R, false, false);
        }
        // D layout: reg r -> element idx = 16*(r + 8*hh) + nn
        const int col = (wid == 0) ? 1 : 2;
        #pragma unroll
        for (int r = 0; r < 8; ++r) {
            const int idx = 16 * (r + 8 * hh) + nn;
            const float val = (idx == 0) ? 0.f
                                         : R[r] * (255.f / (float)idx);
            X[(size_t)(n0 + idx) * 3 + col] = val;
        }
    }
}

// ---------------------------------------------------------------------------
// Kernel 2: one edge per thread. Graph id = src>>8 (T=256 contiguous blocks).
// E rows staged in LDS so the [E,5] stream is written fully coalesced.
// ---------------------------------------------------------------------------
__global__ __launch_bounds__(256) void traj_edge_kernel(
    const float* __restrict__ P,
    const int*   __restrict__ src,
    const int*   __restrict__ tgt,
    const float* __restrict__ X,
    const float* __restrict__ posstd,
    float* __restrict__ E,
    int ne)
{
    __shared__ float stage[256 * 5];

    const int e0 = blockIdx.x * 256;
    const int e  = e0 + threadIdx.x;

    if (e < ne) {
        const int s  = src[e];
        const int tg = tgt[e];
        const int g  = s >> 8;
        const int ts = s & 255;
        const int tt = tg & 255;

        const float td   = (float)(tt - ts) * (1.f / 255.f); // > 0
        const float ratd = 1.f / td;                          // atd == td
        const float sf   = posstd[g];
        const float inv  = 1.f / sf;

        const float psx = P[3 * (size_t)s + 0];
        const float psy = P[3 * (size_t)s + 1];
        const float psz = P[3 * (size_t)s + 2];
        const float ptx = P[3 * (size_t)tg + 0];
        const float pty = P[3 * (size_t)tg + 1];
        const float ptz = P[3 * (size_t)tg + 2];

        const float dx = (ptx - psx) * inv;
        const float dy = (pty - psy) * inv;
        const float dz = (ptz - psz) * inv;
        const float dEu = sqrtf(dx * dx + dy * dy + dz * dz);

        // source diff always exists (ts <= 254 by construction)
        const float dsx = P[3 * (size_t)(s + 1) + 0] - psx;
        const float dsy = P[3 * (size_t)(s + 1) + 1] - psy;
        const float dsz = P[3 * (size_t)(s + 1) + 2] - psz;
        float dtx = 0.f, dty = 0.f, dtz = 0.f;
        if (tt < 255) {
            dtx = P[3 * (size_t)(tg + 1) + 0] - ptx;
            dty = P[3 * (size_t)(tg + 1) + 1] - pty;
            dtz = P[3 * (size_t)(tg + 1) + 2] - ptz;
        }
        // corr = dot(dr_s[tgt]*L, dr_s[src]*L), L = 256
        const float corr = (dsx * dtx + dsy * dty + dsz * dtz)
                           * inv * inv * 65536.f;

        const float cds = X[3 * (size_t)s  + 1];
        const float css = X[3 * (size_t)s  + 2];
        const float cdt = X[3 * (size_t)tg + 1];
        const float cst = X[3 * (size_t)tg + 2];

        const int b = threadIdx.x * 5;
        stage[b + 0] = td;
        stage[b + 1] = dEu * ratd;
        stage[b + 2] = corr;
        stage[b + 3] = (cdt - cds) * ratd;
        stage[b + 4] = (cst - css) * ratd;
    }
    __syncthreads();

    const int nrow  = (ne - e0 < 256) ? (ne - e0) : 256;
    const int nvals = nrow * 5;
    for (int i = threadIdx.x; i < nvals; i += 256)
        E[(size_t)e0 * 5 + i] = stage[i];
}

// ---------------------------------------------------------------------------
extern "C" void kernel_launch(void* const* d_in, const int* in_sizes, int n_in,
                              void* d_out, int out_size, void* d_ws, size_t ws_size,
                              hipStream_t stream)
{
    const float* P   = (const float*)d_in[0];
    // d_in[1] = B (graph ids): structure is implied by T=256, unused
    const int*   src = (const int*)d_in[2];
    const int*   tgt = (const int*)d_in[3];

    const int N  = in_sizes[0] / 3;
    const int Gn = N / TPTS;
    const int ne = in_sizes[2];

    float* X      = (float*)d_out;
    float* E      = X + (size_t)N * 3;
    float* scales = E + (size_t)ne * 5;
    float* posstd = (float*)d_ws;   // G floats of scratch

    traj_node_kernel<<<Gn, 256, 0, stream>>>(P, X, scales, posstd, Gn);
    traj_edge_kernel<<<(ne + 255) / 256, 256, 0, stream>>>(
        P, src, tgt, X, posstd, E, ne);
}